// RotateRoIAlign_16020228014579
// MI455X (gfx1250) — compile-verified
//
#include <hip/hip_runtime.h>
#include <cstdint>
#include <cstddef>

// Problem constants (fixed by the reference harness).
constexpr int kB  = 2;
constexpr int kC  = 256;
constexpr int kH  = 200;
constexpr int kW  = 200;
constexpr int kHW = kH * kW;
constexpr int kN  = 512;
constexpr int kPH = 7;
constexpr int kPW = 7;
constexpr int kBins = kPH * kPW;
constexpr float kScale = 0.25f;

// ---------------------------------------------------------------------------
// gfx1250 async global->LDS copy (ASYNCcnt-tracked). Falls back to a plain
// load+store if the builtin is unavailable on this toolchain.
// Builtin prototype (probe-confirmed): (AS1 int*, AS3 int*, Imm, Imm).
// ---------------------------------------------------------------------------
__device__ __forceinline__ void async_copy_g2l_b32(const float* g, float* l) {
#if defined(__gfx1250__) && __has_builtin(__builtin_amdgcn_global_load_async_to_lds_b32)
  __builtin_amdgcn_global_load_async_to_lds_b32(
      (__attribute__((address_space(1))) int*)(g),
      (__attribute__((address_space(3))) int*)(l),
      /*offset=*/0, /*cpol=*/0);
#else
  *l = *g;
#endif
}

__device__ __forceinline__ void async_wait_all() {
#if defined(__gfx1250__) && __has_builtin(__builtin_amdgcn_global_load_async_to_lds_b32)
#if __has_builtin(__builtin_amdgcn_s_wait_asynccnt)
  __builtin_amdgcn_s_wait_asynccnt(0);
#else
  asm volatile("s_wait_asynccnt 0" ::: "memory");
#endif
#endif
}

// ---------------------------------------------------------------------------
// Kernel 1: NCHW -> NHWC transpose of the feature map.
// Treats input as (B, C, HW) and produces (B, HW, C). 32x32 tiles through a
// padded LDS buffer; global->LDS leg uses async-to-LDS (ASYNCcnt).
// ---------------------------------------------------------------------------
__global__ __launch_bounds__(256) void transpose_nchw_to_nhwc(
    const float* __restrict__ in, float* __restrict__ out) {
  __shared__ float tile[32][33];  // +1 pad: conflict-free transposed reads
  const int tx = threadIdx.x;     // 0..31
  const int ty = threadIdx.y;     // 0..7
  const int b  = blockIdx.z;
  const int c0 = blockIdx.y * 32;   // channel tile
  const int s0 = blockIdx.x * 32;   // spatial (h*w) tile

  const float* src = in + ((size_t)b * kC + c0) * kHW + (size_t)s0 + tx;
#pragma unroll
  for (int r = ty; r < 32; r += 8) {
    async_copy_g2l_b32(src + (size_t)r * kHW, &tile[r][tx]);
  }
  async_wait_all();
  __syncthreads();

  float* dst = out + ((size_t)b * kHW + s0) * kC + (size_t)c0 + tx;
#pragma unroll
  for (int r = ty; r < 32; r += 8) {
    dst[(size_t)r * kC] = tile[tx][r];
  }
}

// ---------------------------------------------------------------------------
// Kernel 2: rotated ROI align. One block per roi; threadIdx.x = channel.
// NHWC=true  : feat is (B, H, W, C) -> corner reads are 1KB contiguous.
// NHWC=false : feat is (B, C, H, W) fallback (no workspace).
// 49-bin loop is software-pipelined (unroll-by-2): offsets for bin k+1 are
// computed and prefetched while bin k's gather FMAs are in flight.
// Results staged in LDS, then flushed with coalesced b128 stores.
// ---------------------------------------------------------------------------
template <bool NHWC>
__global__ __launch_bounds__(256) void rroi_align_kernel(
    const float* __restrict__ feat, const float* __restrict__ rois,
    float* __restrict__ out) {
  __shared__ float res[kBins * kC];  // 49 KB

  const int n = blockIdx.x;
  const int c = threadIdx.x;

  const float* r = rois + (size_t)n * 6;
  const int   b   = (int)r[0];
  const float cx  = r[1] * kScale;
  const float cy  = r[2] * kScale;
  const float rw  = fmaxf(r[3] * kScale, 1.0f);
  const float rh  = fmaxf(r[4] * kScale, 1.0f);
  const float th  = r[5];
  const float ct  = cosf(th);
  const float st  = sinf(th);
  const float bin_h = rh * (1.0f / (float)kPH);
  const float bin_w = rw * (1.0f / (float)kPW);

  // Compute bilinear weights + gather offsets for one bin (16 corners).
  auto computeBin = [&](int bin, float* wgt, unsigned* off) {
    const int ph = bin / kPW;
    const int pw = bin - ph * kPW;
#pragma unroll
    for (int s = 0; s < 4; ++s) {            // 2x2 sub-samples
      const int sy = s >> 1, sx = s & 1;
      const float yy = -0.5f * rh + ((float)ph + ((float)sy + 0.5f) * 0.5f) * bin_h;
      const float xx = -0.5f * rw + ((float)pw + ((float)sx + 0.5f) * 0.5f) * bin_w;
      float y = yy * ct - xx * st + cy;
      float x = yy * st + xx * ct + cx;
      const bool valid = (y > -1.0f) && (y < (float)kH) &&
                         (x > -1.0f) && (x < (float)kW);
      y = fmaxf(y, 0.0f);
      x = fmaxf(x, 0.0f);
      int yl = (int)floorf(y);
      int xl = (int)floorf(x);
      int yh, xh;
      if (yl >= kH - 1) { yl = kH - 1; yh = kH - 1; y = (float)yl; } else { yh = yl + 1; }
      if (xl >= kW - 1) { xl = kW - 1; xh = kW - 1; x = (float)xl; } else { xh = xl + 1; }
      const float ly = y - (float)yl;
      const float lx = x - (float)xl;
      const float hy = 1.0f - ly;
      const float hx = 1.0f - lx;
      const float vm = valid ? 0.25f : 0.0f;  // fold mask + 1/4 mean
      wgt[s * 4 + 0] = vm * hy * hx;
      wgt[s * 4 + 1] = vm * hy * lx;
      wgt[s * 4 + 2] = vm * ly * hx;
      wgt[s * 4 + 3] = vm * ly * lx;
      if (NHWC) {
        const unsigned rowl = (unsigned)(b * kH + yl) * kW;
        const unsigned rowh = (unsigned)(b * kH + yh) * kW;
        off[s * 4 + 0] = (rowl + xl) * kC + c;
        off[s * 4 + 1] = (rowl + xh) * kC + c;
        off[s * 4 + 2] = (rowh + xl) * kC + c;
        off[s * 4 + 3] = (rowh + xh) * kC + c;
      } else {
        const unsigned base = (unsigned)(b * kC + c);
        const unsigned rowl = (base * kH + yl) * kW;
        const unsigned rowh = (base * kH + yh) * kW;
        off[s * 4 + 0] = rowl + xl;
        off[s * 4 + 1] = rowl + xh;
        off[s * 4 + 2] = rowh + xl;
        off[s * 4 + 3] = rowh + xh;
      }
    }
  };

  auto prefetchBin = [&](const unsigned* off) {
#pragma unroll
    for (int k = 0; k < 16; ++k) {
      __builtin_prefetch(feat + off[k], 0, 3);  // global_prefetch_b8
    }
  };

  auto gatherBin = [&](int bin, const float* wgt, const unsigned* off) {
    float acc = 0.0f;
#pragma unroll
    for (int k = 0; k < 16; ++k) {
      acc = fmaf(wgt[k], feat[off[k]], acc);
    }
    // res laid out so the flush below is linear in (c, bin) = output order.
    // 49 and 64 are coprime -> conflict-free LDS banks across a wave.
    res[c * kBins + bin] = acc;
  };

  float    wgtA[16], wgtB[16];
  unsigned offA[16], offB[16];

  computeBin(0, wgtA, offA);
  prefetchBin(offA);
  for (int bin = 0; bin + 1 < kBins; bin += 2) {
    computeBin(bin + 1, wgtB, offB);   // overlap with A's loads
    prefetchBin(offB);
    gatherBin(bin, wgtA, offA);
    if (bin + 2 < kBins) {
      computeBin(bin + 2, wgtA, offA); // overlap with B's loads
      prefetchBin(offA);
    }
    gatherBin(bin + 1, wgtB, offB);
  }
  gatherBin(kBins - 1, wgtA, offA);    // kBins is odd -> tail bin from A

  __syncthreads();

  // Fully coalesced contiguous 50 KB flush per roi, b128 granularity.
  const float4* res4 = (const float4*)res;
  float4* out4 = (float4*)(out + (size_t)n * (kC * kBins));
  constexpr int kVec = (kC * kBins) / 4;  // 3136 float4s
  for (int i = c; i < kVec; i += 256) {
    out4[i] = res4[i];
  }
}

// ---------------------------------------------------------------------------
// Launcher
// ---------------------------------------------------------------------------
extern "C" void kernel_launch(void* const* d_in, const int* in_sizes, int n_in,
                              void* d_out, int out_size, void* d_ws, size_t ws_size,
                              hipStream_t stream) {
  (void)in_sizes; (void)n_in; (void)out_size;
  const float* inputs = (const float*)d_in[0];  // (B, C, H, W) fp32
  const float* rois   = (const float*)d_in[1];  // (N, 6) fp32
  float* out = (float*)d_out;                   // (N, C, PH, PW) fp32

  const size_t nhwcBytes = (size_t)kB * kC * kHW * sizeof(float);  // ~82 MB
  if (d_ws != nullptr && ws_size >= nhwcBytes) {
    float* nhwc = (float*)d_ws;
    dim3 tb(32, 8, 1);
    dim3 tg(kHW / 32, kC / 32, kB);
    transpose_nchw_to_nhwc<<<tg, tb, 0, stream>>>(inputs, nhwc);
    rroi_align_kernel<true><<<kN, 256, 0, stream>>>(nhwc, rois, out);
  } else {
    rroi_align_kernel<false><<<kN, 256, 0, stream>>>(inputs, rois, out);
  }
}